// UpSampler_15925738734010
// MI455X (gfx1250) — compile-verified
//
#include <hip/hip_runtime.h>

// Problem constants (fixed by the reference harness)
#define B_   4
#define F_   128
#define N1_  4096
#define N2_  16384
#define K_   8
#define EPS_ 1e-7f

typedef __attribute__((ext_vector_type(2))) float v2f;
typedef __attribute__((ext_vector_type(8))) float v8f;

// ---------------------------------------------------------------------------
// Kernel 1: KNN (k=8) via V_WMMA_F32_16X16X4_F32 + per-lane sorted top-8,
// then IDW weight computation. One wave handles 16 queries; 8 waves/block.
//
// A (16x4, M=sources): row m = (-2sx, -2sy, -2sz, |s|^2)
// B (4x16, N=queries): col n = ( qx,   qy,   qz,  1    )
// D[m,n] = |s_m|^2 - 2 s_m.q_n   (= d^2 - |q_n|^2, constant shift per query)
//
// 2x-unrolled software pipeline with ping-pong accumulators: each WMMA
// overlaps the other buffer's top-8 filter, and no cross-iteration register
// copies are needed.
// ---------------------------------------------------------------------------
__global__ __launch_bounds__(256) void knn_idw_kernel(
    const float* __restrict__ xyz,   // (B, N1, 3)
    const float* __restrict__ xyzq,  // (B, N2, 3)
    int*   __restrict__ out_idx,     // (B, N2, 8)
    float* __restrict__ out_w)       // (B, N2, 8) normalized IDW weights
{
    __shared__ float4 s_src[N1_];    // 64 KB: (-2x, -2y, -2z, |s|^2)

    const int tid  = threadIdx.x;
    const int lane = tid & 31;
    const int wave = tid >> 5;
    const int n    = lane & 15;      // query column within tile
    const int half = lane >> 4;      // 0: K=0,1 / M=0..7 ; 1: K=2,3 / M=8..15

    const int qblocks = N2_ / 128;   // 128 queries per block (8 waves x 16)
    const int b  = blockIdx.x / qblocks;
    const int q0 = (blockIdx.x % qblocks) * 128;

    // Stage all source points of this batch into LDS (once per block).
    const float* sp = xyz + (size_t)b * (N1_ * 3);
    for (int e = tid; e < N1_; e += 256) {
        float sx = sp[e * 3 + 0];
        float sy = sp[e * 3 + 1];
        float sz = sp[e * 3 + 2];
        s_src[e] = make_float4(-2.f * sx, -2.f * sy, -2.f * sz,
                               sx * sx + sy * sy + sz * sz);
    }
    __syncthreads();

    // Load this lane's query, build the (loop-invariant) B operand.
    const int qi = q0 + wave * 16 + n;
    const float qx = xyzq[((size_t)b * N2_ + qi) * 3 + 0];
    const float qy = xyzq[((size_t)b * N2_ + qi) * 3 + 1];
    const float qz = xyzq[((size_t)b * N2_ + qi) * 3 + 2];
    const float qq = qx * qx + qy * qy + qz * qz;
    v2f bop;
    bop.x = half ? qz   : qx;   // K=0 (lanes 0-15) / K=2 (lanes 16-31)
    bop.y = half ? 1.0f : qy;   // K=1              / K=3

    // Per-lane top-8 (sorted descending: best[0] = current worst kept value).
    float best[K_];
    int   bidx[K_];
#pragma unroll
    for (int j = 0; j < K_; ++j) { best[j] = 3.0e38f; bidx[j] = 0; }

    // One 16x16 distance tile via WMMA.
    auto tile_wmma = [&](int t) -> v8f {
        float4 s4 = s_src[t * 16 + n];
        v2f aop;
        aop.x = half ? s4.z : s4.x;
        aop.y = half ? s4.w : s4.y;
        v8f c = {};
        return __builtin_amdgcn_wmma_f32_16x16x4_f32(
            false, aop, false, bop, (short)0, c, false, false);
    };

    // Top-8 filter for one computed tile.
    auto tile_filter = [&](const v8f& d, int t) {
        // Wave-level whole-tile reject: min over the 8 candidates per lane.
        float m0   = fminf(d[0], d[1]);
        float m1   = fminf(d[2], d[3]);
        float m2   = fminf(d[4], d[5]);
        float m3   = fminf(d[6], d[7]);
        float tmin = fminf(fminf(m0, m1), fminf(m2, m3));
        if (!__builtin_amdgcn_ballot_w32(tmin < best[0]))
            return;                                   // hot fast path
#pragma unroll
        for (int r = 0; r < 8; ++r) {
            float val = d[r];
            int   src = t * 16 + half * 8 + r;
            if (__builtin_amdgcn_ballot_w32(val < best[0])) {
                bool take = val < best[0];
                best[0] = take ? val : best[0];
                bidx[0] = take ? src : bidx[0];
#pragma unroll
                for (int j = 0; j < K_ - 1; ++j) {    // re-sift (one pass)
                    bool s = best[j] < best[j + 1];
                    float tv = best[j]; int ti = bidx[j];
                    best[j]     = s ? best[j + 1] : best[j];
                    bidx[j]     = s ? bidx[j + 1] : bidx[j];
                    best[j + 1] = s ? tv : best[j + 1];
                    bidx[j + 1] = s ? ti : bidx[j + 1];
                }
            }
        }
    };

    // Ping-pong software pipeline (NT is even: 256).
    const int NT = N1_ / 16;
    v8f dA = tile_wmma(0);
    v8f dB = tile_wmma(1);
    int t = 0;
    for (; t < NT - 2; t += 2) {
        tile_filter(dA, t);          // overlaps wmma producing next dB
        dA = tile_wmma(t + 2);
        tile_filter(dB, t + 1);      // overlaps wmma producing next dA
        dB = tile_wmma(t + 3);
    }
    tile_filter(dA, NT - 2);
    tile_filter(dB, NT - 1);

    // Merge the two half-lane candidate sets: lane n absorbs lane n+16's 8.
#pragma unroll
    for (int j = 0; j < K_; ++j) {
        float ov = __shfl_xor(best[j], 16, 32);
        int   oi = __shfl_xor(bidx[j], 16, 32);
        if (half == 0) {
            bool take = ov < best[0];
            best[0] = take ? ov : best[0];
            bidx[0] = take ? oi : bidx[0];
#pragma unroll
            for (int m = 0; m < K_ - 1; ++m) {
                bool s = best[m] < best[m + 1];
                float tv = best[m]; int ti = bidx[m];
                best[m]     = s ? best[m + 1] : best[m];
                bidx[m]     = s ? bidx[m + 1] : bidx[m];
                best[m + 1] = s ? tv : best[m + 1];
                bidx[m + 1] = s ? ti : bidx[m + 1];
            }
        }
    }

    if (half == 0) {
        // Recover true distances (add back |q|^2), compute normalized weights.
        float w[K_];
        float wsum = 0.f;
#pragma unroll
        for (int j = 0; j < K_; ++j) {
            float d2   = fmaxf(best[j] + qq, 0.f);
            float dist = __builtin_sqrtf(d2);
            w[j] = (1.0f + EPS_) / (dist + EPS_);   // DIST_POWER = 1
            wsum += w[j];
        }
        float inv  = 1.0f / wsum;
        int   base = ((size_t)b * N2_ + qi) * K_;
#pragma unroll
        for (int j = 0; j < K_; ++j) {
            out_idx[base + j] = bidx[j];
            out_w  [base + j] = w[j] * inv;
        }
    }
}

// ---------------------------------------------------------------------------
// Kernel 2: gather neighbor features (L2-resident) + weighted sum.
// One thread per (b, n2); loops over all F channels, coalesced output.
// ---------------------------------------------------------------------------
__global__ __launch_bounds__(256) void idw_gather_kernel(
    const float* __restrict__ feat,   // (B, F, N1)
    const int*   __restrict__ nidx,   // (B, N2, 8)
    const float* __restrict__ nw,     // (B, N2, 8)
    float*       __restrict__ out)    // (B, F, N2)
{
    int q  = blockIdx.x * 256 + threadIdx.x;
    int b  = q / N2_;
    int n2 = q % N2_;

    // Vectorized idx/weight loads.
    const int4*   ip = (const int4*)  (nidx + (size_t)q * K_);
    const float4* wp = (const float4*)(nw   + (size_t)q * K_);
    int4   i0 = ip[0], i1 = ip[1];
    float4 w0 = wp[0], w1 = wp[1];
    int   idx[K_] = { i0.x, i0.y, i0.z, i0.w, i1.x, i1.y, i1.z, i1.w };
    float w  [K_] = { w0.x, w0.y, w0.z, w0.w, w1.x, w1.y, w1.z, w1.w };

    const float* fb = feat + (size_t)b * (F_ * N1_);
    float*       ob = out  + (size_t)b * (F_ * N2_) + n2;

#pragma unroll 4
    for (int f = 0; f < F_; ++f) {
        const float* fr = fb + (size_t)f * N1_;
        float acc = 0.f;
#pragma unroll
        for (int j = 0; j < K_; ++j)
            acc = fmaf(w[j], fr[idx[j]], acc);
        ob[(size_t)f * N2_] = acc;
    }
}

// ---------------------------------------------------------------------------
extern "C" void kernel_launch(void* const* d_in, const int* in_sizes, int n_in,
                              void* d_out, int out_size, void* d_ws, size_t ws_size,
                              hipStream_t stream) {
    (void)in_sizes; (void)n_in; (void)out_size; (void)ws_size;

    const float* features = (const float*)d_in[0];  // (B, F, N1, 1)
    const float* xyz      = (const float*)d_in[1];  // (B, N1, 3)
    const float* xyzq     = (const float*)d_in[2];  // (B, N2, 3)
    float*       out      = (float*)d_out;          // (B, F, N2, 1)

    // Workspace layout: [ int idx (B*N2*8) | float w (B*N2*8) ]  = 4 MB total
    int*   ws_idx = (int*)d_ws;
    float* ws_w   = (float*)((char*)d_ws + (size_t)B_ * N2_ * K_ * sizeof(int));

    knn_idw_kernel<<<(B_ * N2_) / 128, 256, 0, stream>>>(xyz, xyzq, ws_idx, ws_w);
    idw_gather_kernel<<<(B_ * N2_) / 256, 256, 0, stream>>>(features, ws_idx, ws_w, out);
}